// GFRTModel_31834297598230
// MI455X (gfx1250) — compile-verified
//
#include <hip/hip_runtime.h>
#include <hip/hip_bf16.h>
#include <math.h>

typedef __attribute__((ext_vector_type(2))) float v2f;
typedef __attribute__((ext_vector_type(8))) float v8f;

// ---------------------------------------------------------------------------
// Zero a float buffer (workspace is poisoned; we must init accumulators).
// ---------------------------------------------------------------------------
__global__ void zero_f32(float* __restrict__ p, size_t n) {
    size_t i = (size_t)blockIdx.x * blockDim.x + threadIdx.x;
    size_t stride = (size_t)gridDim.x * blockDim.x;
    for (; i < n; i += stride) p[i] = 0.0f;
}

// ---------------------------------------------------------------------------
// wsum[n] = sum of ee_weight over edges with ee_src == n  (layer-invariant)
// ---------------------------------------------------------------------------
__global__ void ee_wsum_kernel(const int* __restrict__ src,
                               const float* __restrict__ w,
                               float* __restrict__ wsum, int n_edges) {
    int i = blockIdx.x * blockDim.x + threadIdx.x;
    int stride = gridDim.x * blockDim.x;
    for (; i < n_edges; i += stride) atomicAdd(&wsum[src[i]], w[i]);
}

// ---------------------------------------------------------------------------
// Pack fused weight matrix Wcat (64 x 256) = [Wa_top | Wa_bot | Wn | Ws]
// and fused bias (256) = [ba | 0 | bn | bs] for one layer.
// ---------------------------------------------------------------------------
__global__ void pack_weights(const float* __restrict__ Wa,
                             const float* __restrict__ Wn,
                             const float* __restrict__ Ws,
                             const float* __restrict__ ba,
                             const float* __restrict__ bn,
                             const float* __restrict__ bs,
                             int layer,
                             float* __restrict__ Wcat,
                             float* __restrict__ bias) {
    int idx = blockIdx.x * blockDim.x + threadIdx.x;
    if (idx >= 64 * 256) return;
    int k = idx >> 8;        // 0..63  (input feature)
    int c = idx & 255;       // 0..255 (fused output column)
    float v;
    if (c < 64)        v = Wa[layer * 128 * 64 + k * 64 + c];               // A1 (r side)
    else if (c < 128)  v = Wa[layer * 128 * 64 + (64 + k) * 64 + (c - 64)]; // A2 (h side)
    else if (c < 192)  v = Wn[layer * 64 * 64 + k * 64 + (c - 128)];        // M
    else               v = Ws[layer * 64 * 64 + k * 64 + (c - 192)];        // S
    Wcat[idx] = v;
    if (k == 0) {
        float bv;
        if (c < 64)       bv = ba[layer * 64 + c];        // ba folded into A1
        else if (c < 128) bv = 0.0f;                      // A2 unbiased
        else if (c < 192) bv = bn[layer * 64 + (c - 128)];
        else              bv = bs[layer * 64 + (c - 192)];
        bias[c] = bv;
    }
}

// ---------------------------------------------------------------------------
// Node-projection GEMM:  P (Npad x 256) = x (N x 64) @ Wcat (64 x 256) + bias
// - Wcat staged in LDS once per block (coalesced float4), conflict-free
//   stride 264: B-read bank = (8*k0 + col) & 63 -> lanes 0..15 and 16..31
//   occupy disjoint 16-bank ranges.
// - One wave per 16-row strip; sweeps 16 column tiles; K=64 as 16 chained
//   V_WMMA_F32_16X16X4_F32 per tile (ISA 7.12.2 fp32 layouts).
// - P is padded to a multiple of 16 rows -> unconditional stores (no exec
//   mask ladders in the hot loop).
// ---------------------------------------------------------------------------
#define WSTRIDE 264

__global__ void __launch_bounds__(256)
gemm_node_proj(const float* __restrict__ x,
               const float* __restrict__ Wcat,
               const float* __restrict__ bias,
               float* __restrict__ P, int n_nodes) {
    __shared__ float sW[64 * WSTRIDE];
    __shared__ float sBias[256];

    const int tid = threadIdx.x;

    // Cooperative LDS stage: 16384 floats as 4096 float4s, 16 per thread.
    for (int i = tid; i < (64 * 256) / 4; i += 256) {
        float4 v = ((const float4*)Wcat)[i];
        int k = (i * 4) >> 8;
        int c = (i * 4) & 255;
        *(float4*)&sW[k * WSTRIDE + c] = v;
    }
    sBias[tid] = bias[tid];
    __syncthreads();

    const int wave = (blockIdx.x * blockDim.x + threadIdx.x) >> 5;
    const int lane = threadIdx.x & 31;
    const int n_row_tiles = (n_nodes + 15) >> 4;
    if (wave >= n_row_tiles) return;           // wave-uniform exit (post-barrier)

    const int rt   = wave;
    const int mrow = lane & 15;
    const int hi   = lane >> 4;                // 0 or 1
    const int arow = rt * 16 + mrow;

    // Load this wave's A strip (16x64): 16 chunks of K=4, two K per lane.
    v2f a[16];
    if (arow < n_nodes) {
        const float* xr = x + (size_t)arow * 64;
        #pragma unroll
        for (int kc = 0; kc < 16; ++kc) {
            int k0 = kc * 4 + hi * 2;
            a[kc].x = xr[k0];
            a[kc].y = xr[k0 + 1];
        }
    } else {
        #pragma unroll
        for (int kc = 0; kc < 16; ++kc) { a[kc].x = 0.0f; a[kc].y = 0.0f; }
    }

    for (int ct = 0; ct < 16; ++ct) {
        const int col = ct * 16 + mrow;
        // C initialized from fused bias (same value for every M row).
        float bv = sBias[col];
        v8f c;
        #pragma unroll
        for (int i = 0; i < 8; ++i) c[i] = bv;

        #pragma unroll
        for (int kc = 0; kc < 16; ++kc) {
            int k0 = kc * 4 + hi * 2;
            v2f b;
            b.x = sW[k0 * WSTRIDE + col];
            b.y = sW[(k0 + 1) * WSTRIDE + col];
            // D = A*B + C   (f32, 16x16x4)
            c = __builtin_amdgcn_wmma_f32_16x16x4_f32(
                    false, a[kc], false, b, (short)0, c, false, false);
        }

        // P is padded to n_row_tiles*16 rows: store unconditionally.
        #pragma unroll
        for (int v = 0; v < 8; ++v) {
            int row = rt * 16 + v + hi * 8;
            P[(size_t)row * 256 + col] = c[v];
        }
    }
}

// ---------------------------------------------------------------------------
// Pass 1: per-edge attention logit.
// e = tanh(A1[dst] + A2[src]) . w0 + b0   (ba already folded into A1)
// segmax[src] = max(0, max e)  via uint atomicMax on non-negative floats.
// One wave per edge; lane handles features {lane, lane+32}.
// ---------------------------------------------------------------------------
__global__ void edge_logits(const float* __restrict__ P,
                            const int* __restrict__ src,
                            const int* __restrict__ dst,
                            const float* __restrict__ w0_all,
                            const float* __restrict__ b0_all,
                            int layer,
                            float* __restrict__ e_buf,
                            unsigned int* __restrict__ segmax_bits,
                            int n_edges) {
    const int lane = threadIdx.x & 31;
    int wave = (blockIdx.x * blockDim.x + threadIdx.x) >> 5;
    const int nwaves = (gridDim.x * blockDim.x) >> 5;
    const float* w0 = w0_all + layer * 64;
    const float b0 = b0_all[layer];
    const float w0a = w0[lane], w0b = w0[lane + 32];

    for (int ed = wave; ed < n_edges; ed += nwaves) {
        int s = src[ed], d = dst[ed];
        const float* a1 = P + (size_t)d * 256;        // A1[dst] (+ba)
        const float* a2 = P + (size_t)s * 256 + 64;   // A2[src]
        float t0 = tanhf(a1[lane] + a2[lane]);
        float t1 = tanhf(a1[lane + 32] + a2[lane + 32]);
        float part = t0 * w0a + t1 * w0b;
        #pragma unroll
        for (int off = 16; off > 0; off >>= 1)
            part += __shfl_xor(part, off, 32);
        float e = part + b0;
        if (lane == 0) {
            e_buf[ed] = e;
            atomicMax(&segmax_bits[s], __float_as_uint(fmaxf(e, 0.0f)));
        }
    }
}

// ---------------------------------------------------------------------------
// Pass 2: expv = exp(e - segmax[src]); accumulate unnormalized message sum
// aggun[src] += expv * M[dst]  and  segsum[src] += expv.
// ---------------------------------------------------------------------------
__global__ void edge_accum(const float* __restrict__ P,
                           const int* __restrict__ src,
                           const int* __restrict__ dst,
                           const float* __restrict__ e_buf,
                           const float* __restrict__ segmax,
                           float* __restrict__ segsum,
                           float* __restrict__ aggun,
                           int n_edges) {
    const int lane = threadIdx.x & 31;
    int wave = (blockIdx.x * blockDim.x + threadIdx.x) >> 5;
    const int nwaves = (gridDim.x * blockDim.x) >> 5;

    for (int ed = wave; ed < n_edges; ed += nwaves) {
        int s = src[ed], d = dst[ed];
        float expv = __expf(e_buf[ed] - segmax[s]);
        if (lane == 0) atomicAdd(&segsum[s], expv);
        const float* M = P + (size_t)d * 256 + 128;   // M[dst] (+bn)
        atomicAdd(&aggun[(size_t)s * 64 + lane],      expv * M[lane]);
        atomicAdd(&aggun[(size_t)s * 64 + lane + 32], expv * M[lane + 32]);
    }
}

// ---------------------------------------------------------------------------
// Pass 3: node update.
// x_new = tanh(S[n] + aggun[n]/(segsum[n]+1e-9) + wsum[n]*M[n])
// ---------------------------------------------------------------------------
__global__ void node_update(const float* __restrict__ P,
                            const float* __restrict__ aggun,
                            const float* __restrict__ segsum,
                            const float* __restrict__ wsum,
                            float* __restrict__ x_out, int n_nodes) {
    const int lane = threadIdx.x & 31;
    int wave = (blockIdx.x * blockDim.x + threadIdx.x) >> 5;
    const int nwaves = (gridDim.x * blockDim.x) >> 5;

    for (int n = wave; n < n_nodes; n += nwaves) {
        float denom = segsum[n] + 1e-9f;
        float w = wsum[n];
        const float* Pm  = P + (size_t)n * 256 + 128;
        const float* Psf = P + (size_t)n * 256 + 192;
        #pragma unroll
        for (int j = 0; j < 2; ++j) {
            int dd = lane + j * 32;
            float val = tanhf(Psf[dd] + aggun[(size_t)n * 64 + dd] / denom + w * Pm[dd]);
            x_out[(size_t)n * 64 + dd] = val;
        }
    }
}

// ---------------------------------------------------------------------------
// Host orchestration
// ---------------------------------------------------------------------------
extern "C" void kernel_launch(void* const* d_in, const int* in_sizes, int n_in,
                              void* d_out, int out_size, void* d_ws, size_t ws_size,
                              hipStream_t stream) {
    const int D = 64;
    const int n_nodes = in_sizes[0] / D;           // 100500
    const int L = in_sizes[3] / D;                 // 2 (w0 is (L,D))
    const int n_rt = (n_nodes + 15) >> 4;
    const int n_pad = n_rt * 16;                   // P row count (padded)
    int emax = in_sizes[9];                        // max er edge count across views
    if (in_sizes[22] > emax) emax = in_sizes[22];

    // Workspace carve-up (floats)
    float* ws = (float*)d_ws;
    size_t off = 0;
    float* x_buf  = ws + off; off += (size_t)n_nodes * D;      // intermediate x
    float* P      = ws + off; off += (size_t)n_pad * 256;      // [A1|A2|M|S] (padded rows)
    float* aggun  = ws + off; off += (size_t)n_nodes * D;      // unnormalized agg
    float* segmax = ws + off; off += (size_t)n_nodes;
    float* segsum = ws + off; off += (size_t)n_nodes;
    float* wsum   = ws + off; off += (size_t)n_nodes;
    float* Wcat   = ws + off; off += 64 * 256;
    float* biasv  = ws + off; off += 256;
    float* e_buf  = ws + off; off += (size_t)emax;

    const int ZBLK = 1024;
    const int EDGE_GRID = 4096;   // grid-stride wave-per-edge kernels

    for (int v = 0; v < 2; ++v) {
        const int b = v * 13;
        const float* node_emb = (const float*)d_in[b + 0];
        const float* Wa  = (const float*)d_in[b + 1];
        const float* ba  = (const float*)d_in[b + 2];
        const float* w0  = (const float*)d_in[b + 3];
        const float* b0  = (const float*)d_in[b + 4];
        const float* Wsf = (const float*)d_in[b + 5];
        const float* bs  = (const float*)d_in[b + 6];
        const float* Wn  = (const float*)d_in[b + 7];
        const float* bn  = (const float*)d_in[b + 8];
        const int*   er_src = (const int*)d_in[b + 9];
        const int*   er_dst = (const int*)d_in[b + 10];
        const int*   ee_src = (const int*)d_in[b + 11];
        const float* ee_w   = (const float*)d_in[b + 12];
        const int E_er = in_sizes[b + 9];
        const int E_ee = in_sizes[b + 11];

        // Layer-invariant ee weight sums
        zero_f32<<<256, ZBLK, 0, stream>>>(wsum, (size_t)n_nodes);
        ee_wsum_kernel<<<(E_ee + 255) / 256, 256, 0, stream>>>(ee_src, ee_w, wsum, E_ee);

        const float* x_in = node_emb;
        for (int l = 0; l < L; ++l) {
            float* x_out = (l == L - 1)
                         ? ((float*)d_out + (size_t)v * n_nodes * D)
                         : x_buf;

            pack_weights<<<64, 256, 0, stream>>>(Wa, Wn, Wsf, ba, bn, bs, l, Wcat, biasv);

            // Zero accumulators (aggun, segmax, segsum are contiguous in ws)
            zero_f32<<<1024, ZBLK, 0, stream>>>(aggun,
                                                (size_t)n_nodes * D + 2 * (size_t)n_nodes);

            // Fused node projection GEMM via f32 WMMA (LDS-staged B)
            int gemm_blocks = (n_rt + 7) / 8;           // 8 waves per 256-thread block
            gemm_node_proj<<<gemm_blocks, 256, 0, stream>>>(x_in, Wcat, biasv, P, n_nodes);

            edge_logits<<<EDGE_GRID, 256, 0, stream>>>(P, er_src, er_dst, w0, b0, l,
                                                       e_buf, (unsigned int*)segmax, E_er);
            edge_accum<<<EDGE_GRID, 256, 0, stream>>>(P, er_src, er_dst, e_buf,
                                                      segmax, segsum, aggun, E_er);
            node_update<<<EDGE_GRID, 256, 0, stream>>>(P, aggun, segsum, wsum,
                                                       x_out, n_nodes);
            x_in = x_out;
        }
    }
}